// TransformerGDLinearApprox_88802743812164
// MI455X (gfx1250) — compile-verified
//
#include <hip/hip_runtime.h>
#include <math.h>

typedef __attribute__((ext_vector_type(2))) float v2f;
typedef __attribute__((ext_vector_type(8))) float v8f;

#define BQ 16
#define NN 2048
#define NCTX 2047
#define DINC 160
#define SPLIT 8

// D = A(16x4) * B(4x16) + C  (fp32 WMMA)
static __device__ __forceinline__ v8f wmma_f32(v2f a, v2f b, v8f c) {
    return __builtin_amdgcn_wmma_f32_16x16x4_f32(false, a, false, b, (short)0, c,
                                                 false, false);
}

// ---------------------------------------------------------------------------
// k1: per (split s, batch b) compute partial U = Xcat_trim*Q_trim^T (64x32)
//     and P = Q_trim*Q_trim^T (32x32), summed over this split's n-chunks.
// Layout facts used (CDNA5 ISA 7.12.2, 32-bit 16x4 A / 4x16 B):
//   A: lane r=lane&15 -> row, k0=2*(lane>>4); VGPR0=k0, VGPR1=k0+1
//   B: lane c=lane&15 -> col, same k mapping
// => both operands are a contiguous float2 load x[b, ch+idx, n0+k0 .. +1].
// ---------------------------------------------------------------------------
__global__ void __launch_bounds__(128)
k1_corr(const float* __restrict__ x, float* __restrict__ partial) {
    const int s = blockIdx.x;            // 0..SPLIT-1
    const int b = blockIdx.y;            // 0..BQ-1
    const int w = threadIdx.x >> 5;      // wave 0..3
    const int lane = threadIdx.x & 31;
    const int r = lane & 15;
    const int k0 = (lane >> 4) * 2;

    v8f acc[12] = {};
    const float* xb = x + (size_t)b * DINC * NN;

    // 512 chunks of 4 columns; this block handles 64 of them, 16 per wave.
    for (int t = 0; t < 16; t++) {
        const int chunk = s * 64 + t * 4 + w;
        const int n0 = chunk * 4;
        v2f op[6];
#pragma unroll
        for (int m = 0; m < 6; m++) {
            // tiles 0,1: xq channels 0..31 ; tiles 2..5: xcat channels 32..95
            const float* p = xb + (size_t)(m * 16 + r) * NN + n0 + k0;
            v2f v; v.x = p[0]; v.y = p[1];
            if (n0 + k0 + 1 == NCTX) v.y = 0.f;   // exclude column n=2047
            op[m] = v;
        }
        // U tiles: rows = xcat (4 tiles), cols = xq (2 tiles)
#pragma unroll
        for (int ti = 0; ti < 4; ti++)
#pragma unroll
            for (int tj = 0; tj < 2; tj++)
                acc[ti * 2 + tj] = wmma_f32(op[2 + ti], op[tj], acc[ti * 2 + tj]);
        // P tiles: rows = xq (2), cols = xq (2)
#pragma unroll
        for (int ti = 0; ti < 2; ti++)
#pragma unroll
            for (int tj = 0; tj < 2; tj++)
                acc[8 + ti * 2 + tj] = wmma_f32(op[ti], op[tj], acc[8 + ti * 2 + tj]);
    }

    // cross-wave reduction in LDS (4 waves x 12 tiles x 256 f32 = 48 KB)
    __shared__ float red[4 * 12 * 256];
#pragma unroll
    for (int tt = 0; tt < 12; tt++)
#pragma unroll
        for (int v = 0; v < 8; v++)
            red[w * 3072 + (tt * 8 + v) * 32 + lane] = acc[tt][v];
    __syncthreads();
    for (int e = threadIdx.x; e < 3072; e += 128) {
        float sum = red[e] + red[e + 3072] + red[e + 2 * 3072] + red[e + 3 * 3072];
        partial[((size_t)(b * SPLIT + s)) * 3072 + e] = sum;
    }
}

// ---------------------------------------------------------------------------
// k1b: reduce SPLIT partials, de-swizzle WMMA C-layout into row-major U, P.
// C layout: vgpr v, lane -> row = v + 8*(lane>>4), col = lane&15.
// ---------------------------------------------------------------------------
__global__ void __launch_bounds__(256)
k1b_reduce(const float* __restrict__ partial, float* __restrict__ U,
           float* __restrict__ P) {
    const int b = blockIdx.x;
    for (int e = threadIdx.x; e < 3072; e += 256) {
        float s = 0.f;
        for (int p = 0; p < SPLIT; p++)
            s += partial[((size_t)(b * SPLIT + p)) * 3072 + e];
        const int t = e >> 8, rr = e & 255, v = rr >> 5, lane = rr & 31;
        const int row = v + 8 * (lane >> 4), col = lane & 15;
        if (t < 8) {
            const int ti = t >> 1, tj = t & 1;
            U[b * 2048 + (ti * 16 + row) * 32 + tj * 16 + col] = s;
        } else {
            const int t2 = t - 8, ti = t2 >> 1, tj = t2 & 1;
            P[b * 1024 + (ti * 16 + row) * 32 + tj * 16 + col] = s;
        }
    }
}

// ---------------------------------------------------------------------------
// k2: per-batch tiny recurrence.
//   S0 = E*U (64x32);  D=0
//   l=0..2:  X = D*P;  G = diag(a_l)*(S0 + M*X);  D += (kp/2047)*G
//   R = E^T * D (64x32)
// ---------------------------------------------------------------------------
__global__ void __launch_bounds__(256)
k2_recur(const float* __restrict__ U, const float* __restrict__ Pm,
         const float* __restrict__ E, const float* __restrict__ M,
         const float* __restrict__ alpha, const float* __restrict__ kpp,
         float* __restrict__ R) {
    const int b = blockIdx.x;
    __shared__ float sU[2048], sP[1024], sS0[2048], sD[2048], sX[2048];
    const float kscale = kpp[0] / (float)NCTX;

    for (int e = threadIdx.x; e < 2048; e += 256) sU[e] = U[b * 2048 + e];
    for (int e = threadIdx.x; e < 1024; e += 256) sP[e] = Pm[b * 1024 + e];
    __syncthreads();

    for (int e = threadIdx.x; e < 2048; e += 256) {
        const int i = e >> 5, j = e & 31;
        float s = 0.f;
        for (int c = 0; c < 64; c++) s += E[i * 64 + c] * sU[c * 32 + j];
        sS0[e] = s;
        sD[e] = 0.f;
    }
    __syncthreads();

    for (int l = 0; l < 3; l++) {
        for (int e = threadIdx.x; e < 2048; e += 256) {
            const int i = e >> 5, j = e & 31;
            float s = 0.f;
            for (int t = 0; t < 32; t++) s += sD[i * 32 + t] * sP[t * 32 + j];
            sX[e] = s;
        }
        __syncthreads();
        for (int e = threadIdx.x; e < 2048; e += 256) {
            const int i = e >> 5, j = e & 31;
            float y = 0.f;
            for (int k = 0; k < 64; k++) y += M[i * 64 + k] * sX[k * 32 + j];
            const float g = alpha[l * 64 + i] * (sS0[e] + y);
            sD[e] += kscale * g;
        }
        __syncthreads();
    }

    for (int e = threadIdx.x; e < 2048; e += 256) {
        const int c = e >> 5, j = e & 31;
        float s = 0.f;
        for (int d = 0; d < 64; d++) s += E[d * 64 + c] * sD[d * 32 + j];
        R[b * 2048 + e] = s;
    }
}

// ---------------------------------------------------------------------------
// k3: logits^T(64c x N) = E^T * x_tail + R * Q  via WMMA, then softmax over c.
// Each wave: 16 columns of n, 4 c-tiles, 24 K-steps (16 for E^T part, 8 for R).
// ---------------------------------------------------------------------------
__global__ void __launch_bounds__(256)
k3_out(const float* __restrict__ x, const float* __restrict__ E,
       const float* __restrict__ R, float* __restrict__ out) {
    const int b = blockIdx.y;
    const int w = threadIdx.x >> 5;
    const int lane = threadIdx.x & 31;
    const int r = lane & 15;
    const int k0 = (lane >> 4) * 2;
    const int nb = blockIdx.x * 128 + w * 16;

    const float* xb = x + (size_t)b * DINC * NN;
    const float* Rb = R + b * 2048;

    v8f acc[4] = {};

    // E^T * x_tail : K = 64 (d)
    for (int d0 = 0; d0 < 64; d0 += 4) {
        v2f bop;
        bop.x = xb[(size_t)(96 + d0 + k0) * NN + nb + r];
        bop.y = xb[(size_t)(96 + d0 + k0 + 1) * NN + nb + r];
#pragma unroll
        for (int ct = 0; ct < 4; ct++) {
            const int c = ct * 16 + r;
            v2f aop;
            aop.x = E[(d0 + k0) * 64 + c];
            aop.y = E[(d0 + k0 + 1) * 64 + c];
            acc[ct] = wmma_f32(aop, bop, acc[ct]);
        }
    }
    // R * Q : K = 32 (j)
    for (int j0 = 0; j0 < 32; j0 += 4) {
        v2f bop;
        bop.x = xb[(size_t)(j0 + k0) * NN + nb + r];
        bop.y = xb[(size_t)(j0 + k0 + 1) * NN + nb + r];
#pragma unroll
        for (int ct = 0; ct < 4; ct++) {
            const int c = ct * 16 + r;
            v2f aop;
            aop.x = Rb[c * 32 + j0 + k0];
            aop.y = Rb[c * 32 + j0 + k0 + 1];
            acc[ct] = wmma_f32(aop, bop, acc[ct]);
        }
    }

    // stage logits in LDS: per wave [c(64)][n_local(16)]
    __shared__ float sL[8 * 1024];
    __shared__ float sMax[8 * 16];
    __shared__ float sInv[8 * 16];
#pragma unroll
    for (int ct = 0; ct < 4; ct++)
#pragma unroll
        for (int v = 0; v < 8; v++) {
            const int c = ct * 16 + v + 8 * (lane >> 4);
            sL[w * 1024 + c * 16 + r] = acc[ct][v];
        }
    __syncthreads();

    if (lane < 16) {
        float m = -3.4e38f;
        for (int c = 0; c < 64; c++) m = fmaxf(m, sL[w * 1024 + c * 16 + lane]);
        float s = 0.f;
        for (int c = 0; c < 64; c++) s += __expf(sL[w * 1024 + c * 16 + lane] - m);
        sMax[w * 16 + lane] = m;
        sInv[w * 16 + lane] = 1.0f / s;
    }
    __syncthreads();

    // coalesced writes: logits then predictions
    for (int it = 0; it < 32; it++) {
        const int idx = it * 32 + lane;        // 0..1023
        const int nl = idx >> 6, c = idx & 63;
        const float v = sL[w * 1024 + c * 16 + nl];
        const float m = sMax[w * 16 + nl];
        const float is = sInv[w * 16 + nl];
        const size_t o = ((size_t)(b * NN) + nb + nl) * 64 + c;
        out[o] = v;
        out[(size_t)BQ * NN * 64 + o] = __expf(v - m) * is;
    }
}

extern "C" void kernel_launch(void* const* d_in, const int* in_sizes, int n_in,
                              void* d_out, int out_size, void* d_ws, size_t ws_size,
                              hipStream_t stream) {
    const float* x     = (const float*)d_in[0];
    const float* alpha = (const float*)d_in[1];
    const float* kp    = (const float*)d_in[2];
    const float* E     = (const float*)d_in[3];
    const float* M     = (const float*)d_in[4];
    float* out = (float*)d_out;
    float* ws  = (float*)d_ws;

    float* partial = ws;                                // 16*8*3072
    float* U = partial + (size_t)BQ * SPLIT * 3072;     // 16*2048
    float* P = U + (size_t)BQ * 2048;                   // 16*1024
    float* R = P + (size_t)BQ * 1024;                   // 16*2048

    k1_corr  <<<dim3(SPLIT, BQ), 128, 0, stream>>>(x, partial);
    k1b_reduce<<<BQ, 256, 0, stream>>>(partial, U, P);
    k2_recur <<<BQ, 256, 0, stream>>>(U, P, E, M, alpha, kp, R);
    k3_out   <<<dim3(NN / 128, BQ), 256, 0, stream>>>(x, E, R, out);
}